// AdaptivePriorBoxesLoss_51393578664011
// MI455X (gfx1250) — compile-verified
//
#include <hip/hip_runtime.h>
#include <hip/hip_bf16.h>
#include <stdint.h>

#define TMAX 128

typedef unsigned int v4u __attribute__((ext_vector_type(4)));
typedef int          v8i __attribute__((ext_vector_type(8)));
typedef int          v4i __attribute__((ext_vector_type(4)));
typedef float        v2f __attribute__((ext_vector_type(2)));
typedef float        v8f __attribute__((ext_vector_type(8)));

// ---------------------------------------------------------------------------
// K0: zero accumulators and per-truth packed (iou,idx) slots
// ---------------------------------------------------------------------------
__global__ void k_init(float* accum, unsigned long long* gBest, int T) {
    int t = threadIdx.x;
    if (t < 4) accum[t] = 0.0f;
    if (t < T) gBest[t] = 0ull;
}

// ---------------------------------------------------------------------------
// K1: one prior per thread; truths tile DMA'd to LDS via the Tensor Data
//     Mover. Produces bto[p] = max_t IoU(t,p) and
//     gBest[t] = packed max_p (IoU, lowest p on ties).
// ---------------------------------------------------------------------------
__global__ void k_main(const float* __restrict__ locs,
                       const float* __restrict__ params,
                       const float* __restrict__ truths,
                       float* __restrict__ bto,
                       unsigned long long* __restrict__ gBest,
                       int P, int T) {
    __shared__ float4 sT[TMAX];
    __shared__ unsigned long long sBest[TMAX];

    if (T > TMAX) T = TMAX;

#if __has_builtin(__builtin_amdgcn_tensor_load_to_lds) && __has_builtin(__builtin_amdgcn_s_wait_tensorcnt)
    if (threadIdx.x == 0) {
        // --- Tensor DMA descriptor (ISA cdna5 §8): 1-D tile of n = T*4 f32 ---
        uint64_t ga   = (uint64_t)truths;
        uint32_t galo = __builtin_amdgcn_readfirstlane((uint32_t)ga);
        uint32_t gahi = __builtin_amdgcn_readfirstlane((uint32_t)(ga >> 32));
        uint32_t lds  = __builtin_amdgcn_readfirstlane((uint32_t)(uintptr_t)&sT[0]);
        uint32_t n    = (uint32_t)(T * 4);           // elements (data_size = 4B)

        v4u g0;
        g0[0] = 1u;                                   // count=1, user mode
        g0[1] = lds;                                  // lds_addr (bytes)
        g0[2] = galo;                                 // global_addr[31:0]
        g0[3] = (gahi & 0x01FFFFFFu) | 0x80000000u;   // global_addr[56:32] | type=2

        v8i g1;
        g1[0] = (int)(2u << 16);                      // wg_mask=0, data_size=4B
        g1[1] = (int)((n & 0xFFFFu) << 16);           // tensor_dim0[15:0] @ bits63:48
        g1[2] = (int)(((n >> 16) & 0xFFFFu) | (1u << 16)); // dim0 hi | tensor_dim1=1
        g1[3] = (int)((n & 0xFFFFu) << 16);           // tile_dim0 = n @ bits127:112
        g1[4] = 0;                                    // tile_dim1=0, tile_dim2=0
        g1[5] = (int)n;                               // tensor_dim0_stride
        g1[6] = 0;
        g1[7] = 0;

        v4i gz4 = {0, 0, 0, 0};
        v8i gz8 = {0, 0, 0, 0, 0, 0, 0, 0};
        __builtin_amdgcn_tensor_load_to_lds(g0, g1, gz4, gz4, gz8, 0);
        __builtin_amdgcn_s_wait_tensorcnt(0);
    }
#else
    for (int t = threadIdx.x; t < T; t += blockDim.x) {
        const float4* tf = (const float4*)truths;
        sT[t] = tf[t];
    }
#endif
    for (int t = threadIdx.x; t < T; t += blockDim.x) sBest[t] = 0ull;
    __syncthreads();

    int p = blockIdx.x * blockDim.x + threadIdx.x;
    bool live = (p < P);
    int pl = live ? p : 0;

    float cx = locs[2 * pl], cy = locs[2 * pl + 1];
    float w  = params[3 * pl], h = params[3 * pl + 1];
    float px0 = cx - 0.5f * w, py0 = cy - 0.5f * h;
    float px1 = cx + 0.5f * w, py1 = cy + 0.5f * h;
    float areaP = w * h;

    float maxv = 0.0f;
    int lane = threadIdx.x & 31;

    for (int t = 0; t < T; ++t) {
        float4 tb = sT[t];
        float iw = fminf(tb.z, px1) - fmaxf(tb.x, px0);
        float ih = fminf(tb.w, py1) - fmaxf(tb.y, py0);
        iw = fmaxf(iw, 0.0f);
        ih = fmaxf(ih, 0.0f);
        float inter = iw * ih;
        float areaT = (tb.z - tb.x) * (tb.w - tb.y);
        float iou = inter / (areaT + areaP - inter);
        if (!live) iou = 0.0f;
        maxv = fmaxf(maxv, iou);

        // wave32 butterfly max (all lanes end with wave max)
        float wm = iou;
        #pragma unroll
        for (int o = 16; o > 0; o >>= 1)
            wm = fmaxf(wm, __shfl_xor(wm, o, 32));

        // elect lowest winning lane; only it touches LDS
        unsigned long long ball = __ballot(iou == wm);
        int winner = (int)__ffsll(ball) - 1;
        if (lane == winner) {
            // pack (iou_bits, ~p): iou>=0 so f32 bits are monotonic; inverted
            // index makes atomicMax prefer the LOWEST prior on IoU ties.
            unsigned long long pk =
                ((unsigned long long)__float_as_uint(wm) << 32) |
                (unsigned long long)(0xFFFFFFFFu - (unsigned)pl);
            atomicMax(&sBest[t], pk);
        }
    }

    if (live) bto[p] = maxv;
    __syncthreads();
    for (int t = threadIdx.x; t < T; t += blockDim.x)
        atomicMax(&gBest[t], sBest[t]);
}

// ---------------------------------------------------------------------------
// K2: scatter overrides: bto[best_idx] = -best_overlap (sign bit marks x=K)
// ---------------------------------------------------------------------------
__global__ void k_apply(const unsigned long long* __restrict__ gBest,
                        float* __restrict__ bto, int T, int P) {
    int t = threadIdx.x;
    if (t < T) {
        unsigned long long pk = gBest[t];
        unsigned idx  = 0xFFFFFFFFu - (unsigned)(pk & 0xFFFFFFFFull);
        unsigned bits = (unsigned)(pk >> 32);
        if (idx < (unsigned)P)
            bto[idx] = __uint_as_float(bits | 0x80000000u);
    }
}

// ---------------------------------------------------------------------------
// K3: final sums: s_num = sum sig*x*log(bto), s_sig = sum sig, s_x = sum x.
//     Block reduction: wave32 shuffles -> 8 partials/quantity in LDS ->
//     one v_wmma_f32_16x16x4_f32 (D = ones(16x4) x B(4x16) = column sums)
//     collapses all 24 partials at once.
// ---------------------------------------------------------------------------
__global__ void k_reduce(const float* __restrict__ params,
                         const float* __restrict__ bto,
                         float* __restrict__ accum, int P) {
    int p = blockIdx.x * blockDim.x + threadIdx.x;
    float s_num = 0.0f, s_sig = 0.0f, s_x = 0.0f;
    if (p < P) {
        float a   = params[3 * p + 2];
        float sig = 1.0f / (1.0f + __expf(-a));
        unsigned bb = __float_as_uint(bto[p]);
        bool  isK = (bb >> 31) != 0u;
        float b   = __uint_as_float(bb & 0x7FFFFFFFu);
        float x   = isK ? 2.5f : (b > 0.4f ? 1.0f : 0.0f);
        s_num = (x > 0.0f) ? sig * x * __logf(b) : 0.0f;
        s_sig = sig;
        s_x   = x;
    }
    #pragma unroll
    for (int o = 16; o > 0; o >>= 1) {
        s_num += __shfl_down(s_num, o, 32);
        s_sig += __shfl_down(s_sig, o, 32);
        s_x   += __shfl_down(s_x,   o, 32);
    }
    __shared__ float r0[8], r1[8], r2[8];
    int wid = threadIdx.x >> 5, lane = threadIdx.x & 31;
    if (lane == 0) { r0[wid] = s_num; r1[wid] = s_sig; r2[wid] = s_x; }
    __syncthreads();

#if __has_builtin(__builtin_amdgcn_wmma_f32_16x16x4_f32)
    // Executed by ALL waves with full EXEC (WMMA requires EXEC == all 1s).
    // B slot (lane, vgpr): column n = lane&15; row k = bijection of
    // (vgpr, lane>>4). Column sums are invariant to row placement, so only
    // n matters. Quantity q occupies columns {2q, 2q+1}; its 8 wave-partials
    // are spread over the 4 rows x 2 columns.
    {
        int n  = lane & 15;
        int hi = lane >> 4;
        v2f a = {1.0f, 1.0f};                  // ones 16x4 A matrix
        v2f b;
        #pragma unroll
        for (int v = 0; v < 2; ++v) {
            float val = 0.0f;
            if (n < 6) {
                int q  = n >> 1;
                int pj = ((v * 2 + hi) << 1) | (n & 1);   // partial 0..7
                val = (q == 0) ? r0[pj] : (q == 1) ? r1[pj] : r2[pj];
            }
            b[v] = val;
        }
        v8f c = {0.0f, 0.0f, 0.0f, 0.0f, 0.0f, 0.0f, 0.0f, 0.0f};
        v8f d = __builtin_amdgcn_wmma_f32_16x16x4_f32(
            false, a, false, b, (short)0, c, false, false);
        float s  = d[0];                        // col (lane&15) sum
        float a0 = __shfl(s, 0, 32) + __shfl(s, 1, 32);
        float a1 = __shfl(s, 2, 32) + __shfl(s, 3, 32);
        float a2 = __shfl(s, 4, 32) + __shfl(s, 5, 32);
        if (threadIdx.x == 0) {
            atomicAdd(&accum[0], a0);
            atomicAdd(&accum[1], a1);
            atomicAdd(&accum[2], a2);
        }
    }
#else
    if (threadIdx.x == 0) {
        float a0 = 0, a1 = 0, a2 = 0;
        #pragma unroll
        for (int i = 0; i < 8; ++i) { a0 += r0[i]; a1 += r1[i]; a2 += r2[i]; }
        atomicAdd(&accum[0], a0);
        atomicAdd(&accum[1], a1);
        atomicAdd(&accum[2], a2);
    }
#endif
}

// ---------------------------------------------------------------------------
// K4: loss = (-s_num + BETA*s_sig) / s_x
// ---------------------------------------------------------------------------
__global__ void k_final(const float* __restrict__ accum, float* __restrict__ out) {
    if (threadIdx.x == 0)
        out[0] = (-accum[0] + 1.0f * accum[1]) / accum[2];
}

// ---------------------------------------------------------------------------
extern "C" void kernel_launch(void* const* d_in, const int* in_sizes, int n_in,
                              void* d_out, int out_size, void* d_ws, size_t ws_size,
                              hipStream_t stream) {
    const float* locs   = (const float*)d_in[0];   // [P,2]
    const float* params = (const float*)d_in[1];   // [P,3]
    const float* truths = (const float*)d_in[2];   // [T,4]
    int P = in_sizes[0] / 2;
    int T = in_sizes[2] / 4;

    char* ws = (char*)d_ws;
    float*               accum = (float*)ws;                       // 16 B
    unsigned long long*  gBest = (unsigned long long*)(ws + 16);   // T*8 B
    float*               bto   = (float*)(ws + 16 + (size_t)TMAX * 8); // P*4 B

    int blocks = (P + 255) / 256;

    k_init  <<<1, 256, 0, stream>>>(accum, gBest, T);
    k_main  <<<blocks, 256, 0, stream>>>(locs, params, truths, bto, gBest, P, T);
    k_apply <<<1, TMAX, 0, stream>>>(gBest, bto, T, P);
    k_reduce<<<blocks, 256, 0, stream>>>(params, bto, accum, P);
    k_final <<<1, 32, 0, stream>>>(accum, (float*)d_out);
}